// CalcAssoc_53145925321404
// MI455X (gfx1250) — compile-verified
//
#include <hip/hip_runtime.h>

// CDNA5 (gfx1250) fused SSN calc-assoc:
//   dot = S^T x P via V_WMMA_F32_16X16X4_F32, then 3x3 neighbor gather.
// B=4, C=64, H=W=256, 16x16 superpixel grid (N=256).

typedef __attribute__((ext_vector_type(2))) float v2f;
typedef __attribute__((ext_vector_type(8))) float v8f;

#define NW 16
#define NH 16
#define NSP 256          // NW*NH
#define CCH 64           // channels
#define HW 65536         // 256*256
#define PXW 16           // pixels per wave (one WMMA column tile)
#define WAVES 2          // waves per block
#define PXB (PXW * WAVES)
#define DOT_STRIDE 20    // padded pixel stride in LDS dot buffer (bank-conflict-free)
#define DOT_PER_WAVE (NSP * DOT_STRIDE)
#define INVALID_DIST 1e16f

__global__ __launch_bounds__(WAVES * 32) void CalcAssoc_wmma_kernel(
    const float* __restrict__ pixel_feats,   // [B, C, H*W]
    const float* __restrict__ spixel_feats,  // [B, C, NSP]
    const int*   __restrict__ index_map,     // [B, H*W]
    float* __restrict__ out)                 // [B, 9, H*W]
{
    extern __shared__ float smem[];
    float* snorm  = smem;                        // [NSP]
    float* pnormp = smem + NSP;                  // [WAVES][2][16]
    float* dotbuf = smem + NSP + WAVES * 32;     // [WAVES][NSP][DOT_STRIDE]

    const int tile = blockIdx.x & ((HW / PXB) - 1);
    const int b    = blockIdx.x / (HW / PXB);
    const int tid  = threadIdx.x;
    const int wave = tid >> 5;
    const int lane = tid & 31;
    const int hi   = lane >> 4;    // half-wave: selects K/M sub-block
    const int m    = lane & 15;    // column / row-within-half

    const float* Pb = pixel_feats  + (size_t)b * CCH * HW;
    const float* Sb = spixel_feats + (size_t)b * CCH * NSP;

    // ---- phase 1: snorm[n] = sum_c S[c][n]^2 (coalesced over n) ----
    for (int n = tid; n < NSP; n += WAVES * 32) {
        float acc = 0.f;
        #pragma unroll 8
        for (int c = 0; c < CCH; ++c) {
            float v = Sb[c * NSP + n];
            acc += v * v;
        }
        snorm[n] = acc;
    }

    // ---- phase 2: B fragments (pixel tile, all K=64) + partial pnorm ----
    // B layout (16x16x4 f32): lane holds N = lane%16, K = v + 2*(lane/16).
    const int px0 = tile * PXB + wave * PXW;
    const int px  = px0 + m;
    float breg0[16], breg1[16];
    float pacc = 0.f;
    #pragma unroll
    for (int kg = 0; kg < 16; ++kg) {
        int c0 = kg * 4 + 2 * hi;
        float v0 = Pb[(size_t)c0 * HW + px];        // coalesced 64B per half-wave
        float v1 = Pb[(size_t)(c0 + 1) * HW + px];
        breg0[kg] = v0; breg1[kg] = v1;
        pacc += v0 * v0 + v1 * v1;
    }
    pnormp[wave * 32 + hi * 16 + m] = pacc;  // halves hold disjoint channel sets

    const int cidx = index_map[(size_t)b * HW + px];
    const int ixc = cidx & (NW - 1);
    const int iyc = cidx >> 4;

    // ---- phase 3: dot[n][px] for all 256 superpixels via WMMA f32 ----
    // A layout (16x4 f32): lane holds M = lane%16 (superpixel), K = v + 2*(lane/16).
    // D layout: lane holds N = lane%16 (pixel), M = v + 8*(lane/16).
    float* wdot = dotbuf + wave * DOT_PER_WAVE;
    #pragma unroll 1
    for (int nt = 0; nt < 16; ++nt) {
        const int n0 = nt * 16;
        v8f acc = {};
        #pragma unroll
        for (int kg = 0; kg < 16; ++kg) {
            int c0 = kg * 4 + 2 * hi;
            v2f a, bb;
            a.x = Sb[(size_t)c0 * NSP + n0 + m];         // S is WGP$-resident (64KB)
            a.y = Sb[(size_t)(c0 + 1) * NSP + n0 + m];
            bb.x = breg0[kg];
            bb.y = breg1[kg];
            acc = __builtin_amdgcn_wmma_f32_16x16x4_f32(
                false, a, false, bb, (short)0, acc, false, false);
        }
        // spill D tile to LDS; stride 20 keeps both half-wave stores conflict-free
        #pragma unroll
        for (int v = 0; v < 8; ++v) {
            int n = n0 + v + 8 * hi;
            wdot[n * DOT_STRIDE + m] = acc[v];
        }
    }

    __syncthreads();

    // ---- phase 4: gather 9 neighbors; halves split k even/odd ----
    const float pn = pnormp[wave * 32 + m] + pnormp[wave * 32 + 16 + m];
    #pragma unroll
    for (int kk = 0; kk < 5; ++kk) {
        int k = kk * 2 + hi;
        if (k <= 8) {
            int dy = k / 3 - 1, dx = k % 3 - 1;
            int nx = ixc + dx, ny = iyc + dy;
            bool valid = (nx >= 0) & (nx < NW) & (ny >= 0) & (ny < NH);
            int nxc = min(max(nx, 0), NW - 1);
            int nyc = min(max(ny, 0), NH - 1);
            int nidx = nyc * NW + nxc;
            float d = pn + snorm[nidx] - 2.f * wdot[nidx * DOT_STRIDE + m];
            out[((size_t)b * 9 + k) * HW + px] = valid ? d : INVALID_DIST;
        }
    }
}

extern "C" void kernel_launch(void* const* d_in, const int* in_sizes, int n_in,
                              void* d_out, int out_size, void* d_ws, size_t ws_size,
                              hipStream_t stream) {
    const float* pixel_feats  = (const float*)d_in[0];
    const float* spixel_feats = (const float*)d_in[1];
    const int*   index_map    = (const int*)d_in[2];
    float* out = (float*)d_out;

    const int B = in_sizes[2] / HW;                      // index_map is [B, H, W]
    const int blocks = B * (HW / PXB);
    const size_t shmem = (size_t)(NSP + WAVES * 32 + WAVES * DOT_PER_WAVE) * sizeof(float);

    CalcAssoc_wmma_kernel<<<blocks, WAVES * 32, shmem, stream>>>(
        pixel_feats, spixel_feats, index_map, out);
}